// FluidEliteZK_27917287424191
// MI455X (gfx1250) — compile-verified
//
#include <hip/hip_runtime.h>
#include <hip/hip_bf16.h>
#include <math.h>

// ---------------- CDNA5 types ----------------
typedef __attribute__((ext_vector_type(2))) float v2f;
typedef __attribute__((ext_vector_type(8))) float v8f;

// ---------------- problem constants ----------------
#define LSITES   16
#define CHI      64
#define PAD      96               // >= max bond dim (94)
#define CORE     (PAD * 2 * PAD)  // padded core: [l<96][d<2][r<96]
#define NP       72               // Jacobi padded size (>= 71)
#define NSWEEP   10
#define TRUNC_EVERY 10

// workspace layout (floats):
//  [0,64)                : vec (head input)
//  [64, 64+16*CORE)      : ctx cores
//  then tmpU (128*PAD), tmpT (CORE)
#define WS_VEC   0
#define WS_CTX   64
#define WS_TMPU  (WS_CTX + LSITES * CORE)
#define WS_TMPT  (WS_TMPU + 128 * PAD)

// =====================================================================
// Kernel A: sequential MPS evolution (single workgroup, 256 threads)
// =====================================================================
__global__ __launch_bounds__(256)
void mps_evolve_kernel(const float* __restrict__ ch,   // cores_hidden (L,1,2,2,1)
                       const float* __restrict__ ci,   // cores_input  (L,1,2,2,1)
                       const float* __restrict__ init, // init_ctx     (L,1,2,1)
                       const int*   __restrict__ toks,
                       int ntok,
                       float* __restrict__ ws)
{
    const int tid = threadIdx.x;
    const int nthr = blockDim.x;
    float* vec  = ws + WS_VEC;
    float* ctx  = ws + WS_CTX;
    float* tmpU = ws + WS_TMPU;
    float* tmpT = ws + WS_TMPT;

    __shared__ float G[NP * NP];
    __shared__ float V[NP * NP];
    __shared__ float lam[NP];
    __shared__ float sig[NP];
    __shared__ int   ordr[NP];
    __shared__ int   pa[NP / 2];
    __shared__ int   qa[NP / 2];
    __shared__ float pc[NP / 2];
    __shared__ float ps[NP / 2];
    __shared__ int   k_sh;

    // zero workspace regions we use
    for (int i = tid; i < LSITES * CORE; i += nthr) ctx[i] = 0.0f;
    if (tid < 64) vec[tid] = 0.0f;
    __syncthreads();
    // initial cores: ctx[s][0][d][0] = init[s][0][d][0]
    if (tid < 2 * LSITES) {
        int s = tid >> 1, d = tid & 1;
        ctx[s * CORE + d * PAD] = init[s * 2 + d];
    }
    __syncthreads();

    int cl[LSITES], cr[LSITES];
#pragma unroll
    for (int s = 0; s < LSITES; ++s) { cl[s] = 1; cr[s] = 1; }

    for (int t = 1; t <= ntok; ++t) {
        const int tok = toks[t - 1];

        // ----- h = Wh*ctx (in place) + direct-sum rank-1 border from Wi*emb -----
        for (int s = 0; s < LSITES; ++s) {
            const float wh00 = ch[s * 4 + 0], wh01 = ch[s * 4 + 1];
            const float wh10 = ch[s * 4 + 2], wh11 = ch[s * 4 + 3];
            const int cla = cl[s], cra = cr[s];
            float* C0 = ctx + s * CORE;
            const int tot = cla * cra;
            for (int p = tid; p < tot; p += nthr) {
                const int l = p / cra, r = p - l * cra;
                float* e0 = C0 + l * 2 * PAD + r;
                const float a0 = e0[0];
                const float a1 = e0[PAD];
                e0[0]   = wh00 * a0 + wh01 * a1;
                e0[PAD] = wh10 * a0 + wh11 * a1;
            }
            if (tid == 0) {
                const int b = (tok >> (15 - s)) & 1;          // emb uses reversed bits
                const float x0 = ci[s * 4 + 0 * 2 + b];
                const float x1 = ci[s * 4 + 1 * 2 + b];
                if (s == 0)               { C0[cra] = x0;                       C0[PAD + cra] = x1; }
                else if (s == LSITES - 1) { C0[cla * 2 * PAD] = x0;             C0[cla * 2 * PAD + PAD] = x1; }
                else                      { C0[cla * 2 * PAD + cra] = x0;       C0[cla * 2 * PAD + PAD + cra] = x1; }
            }
        }
        // bond growth (uniform across threads)
        cr[0] += 1;
#pragma unroll
        for (int s = 1; s < LSITES - 1; ++s) { cl[s] += 1; cr[s] += 1; }
        cl[LSITES - 1] += 1;
        __syncthreads();

        // ----- truncation sweep (fires once, at t = 70) -----
        if ((t % TRUNC_EVERY) == 0 && cr[0] > CHI) {
            for (int s = 0; s < LSITES - 1; ++s) {
                const int cls = cl[s];
                const int n   = cr[s];
                const int m   = cls * 2;                    // A is (m x n), row stride PAD
                const float* A = ctx + s * CORE;

                // Gram G = A^T A   (padded with zeros), V = I
                for (int e = tid; e < NP * NP; e += nthr) {
                    const int c1 = e / NP, c2 = e % NP;
                    float acc = 0.0f;
                    if (c1 < n && c2 < n)
                        for (int row = 0; row < m; ++row)
                            acc += A[row * PAD + c1] * A[row * PAD + c2];
                    G[e] = acc;
                    V[e] = (c1 == c2) ? 1.0f : 0.0f;
                }
                __syncthreads();

                // parallel cyclic Jacobi (round-robin tournament, 36 disjoint pairs)
                for (int sweep = 0; sweep < NSWEEP; ++sweep) {
                    for (int rnd = 0; rnd < NP - 1; ++rnd) {
                        if (tid < NP / 2) {
                            const int i = tid;
                            int p = (i == 0) ? 0 : 1 + ((i - 1 + rnd) % (NP - 1));
                            const int j2 = NP - 1 - i;
                            int q = 1 + ((j2 - 1 + rnd) % (NP - 1));
                            if (p > q) { const int tw = p; p = q; q = tw; }
                            pa[i] = p; qa[i] = q;
                            const float gpp = G[p * NP + p];
                            const float gqq = G[q * NP + q];
                            const float gpq = G[p * NP + q];
                            float c = 1.0f, sn = 0.0f;
                            if (fabsf(gpq) > 1e-20f) {
                                const float tau = (gqq - gpp) / (2.0f * gpq);
                                const float tt  = (tau >= 0.0f ? 1.0f : -1.0f) /
                                                  (fabsf(tau) + sqrtf(1.0f + tau * tau));
                                c  = rsqrtf(1.0f + tt * tt);
                                sn = tt * c;
                            }
                            pc[i] = c; ps[i] = sn;
                        }
                        __syncthreads();
                        // row rotations (pairs touch disjoint rows)
                        for (int e = tid; e < (NP / 2) * NP; e += nthr) {
                            const int pr = e / NP, j = e % NP;
                            const int p = pa[pr], q = qa[pr];
                            const float c = pc[pr], sn = ps[pr];
                            const float gp = G[p * NP + j], gq = G[q * NP + j];
                            G[p * NP + j] = c * gp - sn * gq;
                            G[q * NP + j] = sn * gp + c * gq;
                        }
                        __syncthreads();
                        // column rotations on G and V (disjoint columns)
                        for (int e = tid; e < (NP / 2) * NP; e += nthr) {
                            const int pr = e / NP, j = e % NP;
                            const int p = pa[pr], q = qa[pr];
                            const float c = pc[pr], sn = ps[pr];
                            const float gp = G[j * NP + p], gq = G[j * NP + q];
                            G[j * NP + p] = c * gp - sn * gq;
                            G[j * NP + q] = sn * gp + c * gq;
                            const float vp = V[j * NP + p], vq = V[j * NP + q];
                            V[j * NP + p] = c * vp - sn * vq;
                            V[j * NP + q] = sn * vp + c * vq;
                        }
                        __syncthreads();
                    }
                }

                // sort eigenvalues descending; k = min(CHI, min(m, n))
                if (tid == 0) {
                    for (int j = 0; j < n; ++j) { lam[j] = G[j * NP + j]; ordr[j] = j; }
                    for (int a = 0; a < n; ++a) {
                        int best = a;
                        for (int b = a + 1; b < n; ++b)
                            if (lam[ordr[b]] > lam[ordr[best]]) best = b;
                        const int tw = ordr[a]; ordr[a] = ordr[best]; ordr[best] = tw;
                    }
                    int k = (m < n) ? m : n;
                    if (k > CHI) k = CHI;
                    k_sh = k;
                    for (int j = 0; j < k; ++j) {
                        const float lv = lam[ordr[j]];
                        sig[j] = sqrtf(lv > 0.0f ? lv : 0.0f);
                    }
                }
                __syncthreads();
                const int k = k_sh;

                // U = A * V_sorted * Sigma^-1   -> tmpU (m x k, stride PAD)
                for (int e = tid; e < m * k; e += nthr) {
                    const int row = e / k, j = e - (e / k) * k;
                    const int col = ordr[j];
                    float acc = 0.0f;
                    for (int c2 = 0; c2 < n; ++c2)
                        acc += A[row * PAD + c2] * V[c2 * NP + col];
                    const float sv = sig[j];
                    tmpU[row * PAD + j] = acc / (sv > 1e-30f ? sv : 1e-30f);
                }
                __syncthreads();

                // T[s] <- U (reshape (cls,2,k)); zero removed columns
                {
                    float* C0 = ctx + s * CORE;
                    for (int e = tid; e < m * PAD; e += nthr) {
                        const int row = e / PAD, r = e % PAD;
                        C0[row * PAD + r] = (r < k) ? tmpU[row * PAD + r] : 0.0f;
                    }
                }
                // T[s+1] <- carry @ T[s+1], carry[kk][l] = sig[kk] * V[l][ordr[kk]]
                {
                    float* C1 = ctx + (s + 1) * CORE;
                    const int cr1 = cr[s + 1];
                    for (int e = tid; e < n * 2 * PAD; e += nthr) tmpT[e] = C1[e];
                    __syncthreads();
                    for (int e = tid; e < n * 2 * PAD; e += nthr) {
                        const int row = e / (2 * PAD);
                        const int rem = e % (2 * PAD);
                        const int r   = rem % PAD;
                        float acc = 0.0f;
                        if (row < k && r < cr1) {
                            const int d = rem / PAD;
                            const int col = ordr[row];
                            const float sv = sig[row];
                            for (int l2 = 0; l2 < n; ++l2)
                                acc += sv * V[l2 * NP + col] * tmpT[l2 * 2 * PAD + d * PAD + r];
                        }
                        C1[e] = acc;
                    }
                }
                cr[s] = k;
                cl[s + 1] = k;
                __syncthreads();
            }
        }
    }

    // vec = mean over (l,d) of ctx[8], truncated/padded to CHI
    {
        const float* C8 = ctx + 8 * CORE;
        const int cl8 = cl[8], cr8 = cr[8];
        const int lim = (cr8 < CHI) ? cr8 : CHI;
        for (int r = tid; r < CHI; r += nthr) {
            float acc = 0.0f;
            if (r < lim) {
                for (int l2 = 0; l2 < cl8; ++l2)
                    acc += C8[l2 * 2 * PAD + r] + C8[l2 * 2 * PAD + PAD + r];
                acc /= (float)(cl8 * 2);
            }
            vec[r] = acc;
        }
    }
}

// =====================================================================
// Kernel B: head GEMV (128000 x 64) via V_WMMA_F32_16X16X4_F32
// Each wave32 owns a 16-row tile; B = vec broadcast to all 16 columns,
// so every output column equals head_w_tile @ vec. 16 WMMA steps (K=4).
// Memory bound: 32.8 MB of head_w -> ~1.4 us at 23.3 TB/s.
// =====================================================================
__global__ __launch_bounds__(256)
void head_wmma_kernel(const float* __restrict__ head_w,
                      const float* __restrict__ head_b,
                      const float* __restrict__ vec,
                      float* __restrict__ out,
                      int nrows)
{
    const int lane = threadIdx.x & 31;
    const int wave = threadIdx.x >> 5;
    const int tile = blockIdx.x * (blockDim.x >> 5) + wave;
    const int row_base = tile * 16;
    if (row_base >= nrows) return;            // uniform per wave: EXEC stays all-ones

    const int half = lane >> 4;               // A/B layout: VGPR0 holds K,K+2 ; VGPR1 holds K+1,K+3
    const int r = row_base + (lane & 15);
    const float* wrow = head_w + (size_t)r * CHI;

    v8f acc = {};
#pragma unroll
    for (int k0 = 0; k0 < CHI; k0 += 4) {
        v2f a, b;
        a.x = wrow[k0 + 2 * half];
        a.y = wrow[k0 + 2 * half + 1];
        b.x = vec[k0 + 2 * half];
        b.y = vec[k0 + 2 * half + 1];
        acc = __builtin_amdgcn_wmma_f32_16x16x4_f32(
            /*neg_a=*/false, a, /*neg_b=*/false, b,
            /*c_mod=*/(short)0, acc, /*reuse_a=*/false, /*reuse_b=*/false);
    }

    // D layout: VGPR v -> M=v (lanes 0-15) / M=v+8 (lanes 16-31); columns identical.
    if ((lane & 15) == 0) {
#pragma unroll
        for (int vv = 0; vv < 8; ++vv) {
            const int row = row_base + vv + 8 * half;
            if (row < nrows) out[row] = acc[vv] + head_b[row];
        }
    }
}

// =====================================================================
extern "C" void kernel_launch(void* const* d_in, const int* in_sizes, int n_in,
                              void* d_out, int out_size, void* d_ws, size_t ws_size,
                              hipStream_t stream) {
    const float* cores_hidden = (const float*)d_in[0];
    const float* cores_input  = (const float*)d_in[1];
    const float* head_w       = (const float*)d_in[2];
    const float* head_b       = (const float*)d_in[3];
    const float* init_ctx     = (const float*)d_in[4];
    const int*   token_ids    = (const int*)d_in[5];
    const int ntok = in_sizes[5];
    float* ws = (float*)d_ws;

    mps_evolve_kernel<<<1, 256, 0, stream>>>(cores_hidden, cores_input,
                                             init_ctx, token_ids, ntok, ws);

    const int tiles = (out_size + 15) / 16;
    const int blocks = (tiles + 7) / 8;       // 8 waves per 256-thread block
    head_wmma_kernel<<<blocks, 256, 0, stream>>>(head_w, head_b,
                                                 ws + WS_VEC, (float*)d_out, out_size);
}